// SG_84945863180351
// MI455X (gfx1250) — compile-verified
//
#include <hip/hip_runtime.h>

typedef float v2f __attribute__((ext_vector_type(2)));
typedef float v8f __attribute__((ext_vector_type(8)));

#define NEG 5
#define DIMS 128
#define MM 6
#define TILE 16
#define STRIDE 132              // 128 + 4 floats pad: column reads hit distinct LDS banks
#define NBAGS (TILE * 7)        // 16 word bags + 6*16 ctx bags per block

__global__ __launch_bounds__(128)
void sg_loss_kernel(const int* __restrict__ data,
                    const int* __restrict__ word2morph,
                    const float* __restrict__ w2m_mask,
                    const int* __restrict__ ctx2morph,
                    const float* __restrict__ c2m_mask,
                    const float* __restrict__ emb0,
                    const float* __restrict__ emb1,
                    float* __restrict__ partials,
                    int Btot)
{
    __shared__ float bags[NBAGS * STRIDE];  // rows 0..15: W tile; rows 16+16j: Ctx_j tile
    __shared__ float diag[6 * TILE];
    __shared__ float red[128];

    const int t    = threadIdx.x;
    const int wave = t >> 5;
    const int lane = t & 31;
    const int r0   = blockIdx.x * TILE;

    // ---- Phase 1: masked embedding-bag gathers -> LDS (coalesced b128 per lane) ----
    for (int q = wave; q < NBAGS; q += 4) {
        const float* etab;
        const int*   idxp;
        const float* mskp;
        if (q < TILE) {
            int g = min(r0 + q, Btot - 1);
            size_t o = (size_t)g * MM;
            etab = emb0; idxp = word2morph + o; mskp = w2m_mask + o;
        } else {
            int u = q - TILE;
            int j = u >> 4, row = u & 15;
            int g = min(r0 + row, Btot - 1);
            size_t o = ((size_t)g * (1 + NEG) + j) * MM;
            etab = emb1; idxp = ctx2morph + o; mskp = c2m_mask + o;
        }
        int   id[MM];
        float mk[MM];
        #pragma unroll
        for (int m = 0; m < MM; ++m) { id[m] = idxp[m]; mk[m] = mskp[m]; }
        float4 acc = make_float4(0.f, 0.f, 0.f, 0.f);
        #pragma unroll
        for (int m = 0; m < MM; ++m) {
            const float4 v = ((const float4*)(etab + (size_t)id[m] * DIMS))[lane];
            acc.x = fmaf(mk[m], v.x, acc.x);
            acc.y = fmaf(mk[m], v.y, acc.y);
            acc.z = fmaf(mk[m], v.z, acc.z);
            acc.w = fmaf(mk[m], v.w, acc.w);
        }
        *(float4*)(bags + q * STRIDE + lane * 4) = acc;
    }
    __syncthreads();

    // ---- Phase 2: scores = diag(W x Ctx_j^T) via V_WMMA_F32_16X16X4_F32 ----
    {
        const int n  = lane & 15;          // row of our LDS tiles (M for A, N for B)
        const int kk = (lane >> 4) * 2;    // hi half-wave takes K+2 (per ISA f32 layout)
        const float* ap = bags + n * STRIDE + kk;
        for (int j = wave; j < 6; j += 4) {
            const float* bp = bags + (TILE + j * TILE + n) * STRIDE + kk;
            v8f c = {};
            #pragma unroll
            for (int k0 = 0; k0 < DIMS; k0 += 4) {
                v2f a = *(const v2f*)(ap + k0);
                v2f b = *(const v2f*)(bp + k0);
                c = __builtin_amdgcn_wmma_f32_16x16x4_f32(
                        false, a, false, b, (short)0, c, false, false);
            }
            // diagonal (m,m): lane m holds it in c[m] for m<8; lane m+16 in c[m-8] for m>=8
            const int m7 = lane & 7;
            float dv = c[0];
            dv = (m7 == 1) ? c[1] : dv;
            dv = (m7 == 2) ? c[2] : dv;
            dv = (m7 == 3) ? c[3] : dv;
            dv = (m7 == 4) ? c[4] : dv;
            dv = (m7 == 5) ? c[5] : dv;
            dv = (m7 == 6) ? c[6] : dv;
            dv = (m7 == 7) ? c[7] : dv;
            if (lane < 8)        diag[j * TILE + lane]        = dv;
            else if (lane >= 24) diag[j * TILE + (lane - 16)] = dv;
        }
    }
    __syncthreads();

    // ---- Phase 3: clamped logsigmoid terms + deterministic block reduction ----
    float term = 0.0f;
    if (t < 96) {
        const int jj = t >> 4, row = t & 15;
        const int g = r0 + row;
        if (g < Btot) {
            float s = diag[jj * TILE + row];
            float x = fminf(fmaxf(s, -10.0f), 10.0f);
            if (jj == 0) {
                term = log1pf(__expf(-x));                 // -logsigmoid(clip(pos))
            } else {
                float nm = (float)data[(size_t)g * (2 + 2 * NEG) + 6 + jj];
                term = nm * log1pf(__expf(x));             // -logsigmoid(clip(-neg)) * mask
            }
        }
    }
    red[t] = term;
    __syncthreads();
    #pragma unroll
    for (int off = 64; off > 0; off >>= 1) {
        if (t < off) red[t] += red[t + off];
        __syncthreads();
    }
    if (t == 0) partials[blockIdx.x] = red[0];
}

__global__ __launch_bounds__(256)
void sg_reduce_kernel(const float* __restrict__ partials, float* __restrict__ out, int n)
{
    __shared__ float red[256];
    const int t = threadIdx.x;
    float s = 0.0f;
    for (int i = t; i < n; i += 256) s += partials[i];   // fixed-order strided sum
    red[t] = s;
    __syncthreads();
    #pragma unroll
    for (int off = 128; off > 0; off >>= 1) {
        if (t < off) red[t] += red[t + off];
        __syncthreads();
    }
    if (t == 0) out[0] = red[0];
}

extern "C" void kernel_launch(void* const* d_in, const int* in_sizes, int n_in,
                              void* d_out, int out_size, void* d_ws, size_t ws_size,
                              hipStream_t stream)
{
    const int*   data     = (const int*)  d_in[0];
    const int*   w2m      = (const int*)  d_in[1];
    const float* w2m_mask = (const float*)d_in[2];
    const int*   c2m      = (const int*)  d_in[3];
    const float* c2m_mask = (const float*)d_in[4];
    const float* emb0     = (const float*)d_in[5];
    const float* emb1     = (const float*)d_in[6];

    const int Btot   = in_sizes[1] / MM;            // word2morph is [B, 6]
    const int blocks = (Btot + TILE - 1) / TILE;    // 2048 for B=32768

    float* partials = (float*)d_ws;
    float* out      = (float*)d_out;

    sg_loss_kernel<<<blocks, 128, 0, stream>>>(
        data, w2m, w2m_mask, c2m, c2m_mask, emb0, emb1, partials, Btot);
    sg_reduce_kernel<<<1, 256, 0, stream>>>(partials, out, blocks);
}